// GraphEncoder_74526272520809
// MI455X (gfx1250) — compile-verified
//
#include <hip/hip_runtime.h>

typedef __attribute__((ext_vector_type(2))) float v2f;
typedef __attribute__((ext_vector_type(8))) float v8f;
typedef int v4i __attribute__((vector_size(16)));

#define NODE_DIM 128
#define HIDDEN   256
#define LATENT   128
#define COND     5
#define KSLAB    64

#if __has_builtin(__builtin_amdgcn_global_load_async_to_lds_b128)
#define HAVE_ASYNC_LDS 1
#else
#define HAVE_ASYNC_LDS 0
#endif

__device__ __forceinline__ void wait_async_zero() {
#if __has_builtin(__builtin_amdgcn_s_wait_asynccnt)
    __builtin_amdgcn_s_wait_asynccnt(0);
#else
    asm volatile("s_wait_asynccnt 0" ::: "memory");
#endif
}

// 16-byte global -> LDS copy; async (ASYNCcnt-tracked) when the CDNA5
// builtin is available, plain load+ds_store otherwise.
__device__ __forceinline__ void copy16_to_lds(const float* g, float* l) {
#if HAVE_ASYNC_LDS
    __builtin_amdgcn_global_load_async_to_lds_b128(
        (__attribute__((address_space(1))) v4i*)g,
        (__attribute__((address_space(3))) v4i*)l, 0, 0);
#else
    *reinterpret_cast<float4*>(l) = *reinterpret_cast<const float4*>(g);
#endif
}

// ---------------------------------------------------------------- zero fill
__global__ void k_zero(float* __restrict__ p, long n) {
    long i = (long)blockIdx.x * blockDim.x + threadIdx.x;
    long stride = (long)gridDim.x * blockDim.x;
    for (; i < n; i += stride) p[i] = 0.0f;
}

// ---------------------------------------------------- edge scatter, D = 128
// one edge per wave (32 lanes), 4 contiguous floats per lane
__global__ void k_scatter128(const float* __restrict__ x,
                             const int* __restrict__ src,
                             const int* __restrict__ dst,
                             float* __restrict__ agg, int n_edges) {
    int tid  = blockIdx.x * blockDim.x + threadIdx.x;
    int e    = tid >> 5;
    if (e >= n_edges) return;
    int lane = tid & 31;
    int s = src[e], d = dst[e];
    const float4 v = *reinterpret_cast<const float4*>(x + (long)s * 128 + lane * 4);
    float* out = agg + (long)d * 128 + lane * 4;
    unsafeAtomicAdd(out + 0, v.x);
    unsafeAtomicAdd(out + 1, v.y);
    unsafeAtomicAdd(out + 2, v.z);
    unsafeAtomicAdd(out + 3, v.w);
}

// ---------------------------------------------------- edge scatter, D = 256
__global__ void k_scatter256(const float* __restrict__ x,
                             const int* __restrict__ src,
                             const int* __restrict__ dst,
                             float* __restrict__ agg, int n_edges) {
    int tid  = blockIdx.x * blockDim.x + threadIdx.x;
    int e    = tid >> 5;
    if (e >= n_edges) return;
    int lane = tid & 31;
    int s = src[e], d = dst[e];
    const float4* in = reinterpret_cast<const float4*>(x + (long)s * 256 + lane * 8);
    float4 v0 = in[0];
    float4 v1 = in[1];
    float* out = agg + (long)d * 256 + lane * 8;
    unsafeAtomicAdd(out + 0, v0.x);
    unsafeAtomicAdd(out + 1, v0.y);
    unsafeAtomicAdd(out + 2, v0.z);
    unsafeAtomicAdd(out + 3, v0.w);
    unsafeAtomicAdd(out + 4, v1.x);
    unsafeAtomicAdd(out + 5, v1.y);
    unsafeAtomicAdd(out + 6, v1.z);
    unsafeAtomicAdd(out + 7, v1.w);
}

// -------------------------------------------- fp32 WMMA GEMM + bias + ReLU
// out[M,256] = relu(A[M,K] @ W[K,256] + bias)
// Block = 8 waves: 2 M-tiles x 4 N-tile-groups (4 N-tiles per wave).
// W is staged one 64-row K-slab at a time into LDS (async global->LDS when
// available); B fragments then come from ds_load, A fragments from global.
template <int KVAL>
__global__ __launch_bounds__(256) void k_gemm_relu(const float* __restrict__ A,
                                                   const float* __restrict__ W,
                                                   const float* __restrict__ bias,
                                                   float* __restrict__ out,
                                                   int M) {
    constexpr int N = 256;
    __shared__ float wslab[KSLAB * N];          // 64 KB

    const int tid  = threadIdx.x;
    const int wave = tid >> 5;
    const int lane = tid & 31;
    const int half = lane >> 4;                 // 0: K-pair {0,1}, 1: {2,3}
    const int mr   = lane & 15;                 // A: row M; B: col N
    const int kb   = half * 2;

    const int mtiles = M >> 4;                  // 3125
    int mt = blockIdx.x * 2 + (wave >> 2);
    if (mt >= mtiles) mt = mtiles - 1;          // duplicate last tile (no early
                                                // return: barriers below)
    const int ng = wave & 3;                    // N-tiles [ng*4, ng*4+4)
    const int m0 = mt * 16;

    v8f c[4] = {};
    const float* arow = A + (long)(m0 + mr) * KVAL;

    for (int k0 = 0; k0 < KVAL; k0 += KSLAB) {
        // ---- stage W[k0 .. k0+KSLAB) x N into LDS ----
        const float* gsrc = W + (long)k0 * N;
        #pragma unroll
        for (int i = 0; i < (KSLAB * N / 4) / 256; ++i) {
            int idx = (i * 256 + tid) * 4;
            copy16_to_lds(gsrc + idx, wslab + idx);
        }
#if HAVE_ASYNC_LDS
        wait_async_zero();
#endif
        __syncthreads();

        // ---- accumulate this K-slab ----
        for (int k = 0; k < KSLAB; k += 4) {
            v2f a;
            a.x = arow[k0 + k + kb];
            a.y = arow[k0 + k + kb + 1];
            const float* w0 = wslab + (k + kb)     * N + mr;
            const float* w1 = wslab + (k + kb + 1) * N + mr;
            #pragma unroll
            for (int t = 0; t < 4; ++t) {
                int n0 = (ng * 4 + t) * 16;
                v2f b;
                b.x = w0[n0];
                b.y = w1[n0];
                c[t] = __builtin_amdgcn_wmma_f32_16x16x4_f32(
                           false, a, false, b, (short)0, c[t], false, false);
            }
        }
        __syncthreads();                        // before next slab overwrite
    }

    const int rbase = m0 + half * 8;
    #pragma unroll
    for (int t = 0; t < 4; ++t) {
        int col  = (ng * 4 + t) * 16 + mr;
        float bv = bias[col];
        #pragma unroll
        for (int r = 0; r < 8; ++r) {
            float v = c[t][r] + bv;
            out[(long)(rbase + r) * N + col] = v > 0.0f ? v : 0.0f;
        }
    }
}

// ------------------------------------------------------- column-mean pool
__global__ void k_pool(const float* __restrict__ x, float* __restrict__ pooled, int M) {
    __shared__ float sh[256];
    int col = blockIdx.x;
    float sum = 0.0f;
    for (int r = threadIdx.x; r < M; r += blockDim.x)
        sum += x[(long)r * HIDDEN + col];
    sh[threadIdx.x] = sum;
    __syncthreads();
    for (int off = blockDim.x >> 1; off > 0; off >>= 1) {
        if ((int)threadIdx.x < off) sh[threadIdx.x] += sh[threadIdx.x + off];
        __syncthreads();
    }
    if (threadIdx.x == 0) pooled[col] = sh[0] / (float)M;
}

// ------------------------------------------------------------ dense head
__global__ void k_head(const float* __restrict__ pooled,
                       const float* __restrict__ cond,
                       const float* __restrict__ Wfc, const float* __restrict__ bfc,
                       const float* __restrict__ Wm,  const float* __restrict__ bm,
                       const float* __restrict__ Wl,  const float* __restrict__ bl,
                       float* __restrict__ out) {
    __shared__ float in[HIDDEN + COND];
    __shared__ float h[LATENT];
    int t = threadIdx.x;                        // 0..127
    for (int i = t; i < HIDDEN; i += LATENT) in[i] = pooled[i];
    if (t < COND) in[HIDDEN + t] = cond[t];
    __syncthreads();

    float acc = bfc[t];
    #pragma unroll 1
    for (int i = 0; i < HIDDEN + COND; ++i)
        acc = fmaf(in[i], Wfc[i * LATENT + t], acc);
    h[t] = acc > 0.0f ? acc : 0.0f;
    __syncthreads();

    float zm = bm[t], zl = bl[t];
    #pragma unroll 1
    for (int i = 0; i < LATENT; ++i) {
        zm = fmaf(h[i], Wm[i * LATENT + t], zm);
        zl = fmaf(h[i], Wl[i * LATENT + t], zl);
    }
    out[t]          = zm;
    out[LATENT + t] = zl;
}

// ----------------------------------------------------------------- driver
extern "C" void kernel_launch(void* const* d_in, const int* in_sizes, int n_in,
                              void* d_out, int out_size, void* d_ws, size_t ws_size,
                              hipStream_t stream) {
    const float* nf      = (const float*)d_in[0];
    const float* cond    = (const float*)d_in[1];
    const int*   src     = (const int*)  d_in[2];
    const int*   dst     = (const int*)  d_in[3];
    const float* W1      = (const float*)d_in[4];
    const float* b1      = (const float*)d_in[5];
    const float* W2      = (const float*)d_in[6];
    const float* b2      = (const float*)d_in[7];
    const float* Wfc     = (const float*)d_in[8];
    const float* bfc     = (const float*)d_in[9];
    const float* Wmean   = (const float*)d_in[10];
    const float* bmean   = (const float*)d_in[11];
    const float* Wlogvar = (const float*)d_in[12];
    const float* blogvar = (const float*)d_in[13];

    const int n_edges = in_sizes[2];
    const int M       = in_sizes[0] / NODE_DIM;   // 50000

    // workspace layout (bytes):
    //   agg1   : M*128*4  = 25.6 MB
    //   x1/x2  : M*256*4  = 51.2 MB   (x2 reuses x1: x1 dead after scatter2)
    //   agg2   : M*256*4  = 51.2 MB
    //   pooled : 256*4
    char*  ws     = (char*)d_ws;
    size_t off    = 0;
    float* agg1   = (float*)(ws + off); off += (size_t)M * NODE_DIM * sizeof(float);
    float* x1     = (float*)(ws + off); off += (size_t)M * HIDDEN   * sizeof(float);
    float* agg2   = (float*)(ws + off); off += (size_t)M * HIDDEN   * sizeof(float);
    float* pooled = (float*)(ws + off);
    float* x2     = x1;

    k_zero<<<2048, 256, 0, stream>>>(agg1, (long)M * NODE_DIM);
    k_zero<<<2048, 256, 0, stream>>>(agg2, (long)M * HIDDEN);

    int sthreads = n_edges * 32;
    k_scatter128<<<(sthreads + 255) / 256, 256, 0, stream>>>(nf, src, dst, agg1, n_edges);

    int mtiles  = M / 16;                         // 3125
    int gblocks = (mtiles + 1) / 2;               // 2 M-tiles per block
    k_gemm_relu<NODE_DIM><<<gblocks, 256, 0, stream>>>(agg1, W1, b1, x1, M);

    k_scatter256<<<(sthreads + 255) / 256, 256, 0, stream>>>(x1, src, dst, agg2, n_edges);
    k_gemm_relu<HIDDEN><<<gblocks, 256, 0, stream>>>(agg2, W2, b2, x2, M);

    k_pool<<<HIDDEN, 256, 0, stream>>>(x2, pooled, M);
    k_head<<<1, LATENT, 0, stream>>>(pooled, cond, Wfc, bfc,
                                     Wmean, bmean, Wlogvar, blogvar, (float*)d_out);
}